// Refiner_86268713107543
// MI455X (gfx1250) — compile-verified
//
#include <hip/hip_runtime.h>

typedef float v2f __attribute__((ext_vector_type(2)));
typedef float v8f __attribute__((ext_vector_type(8)));

constexpr int BB   = 64;
constexpr int TT   = 262144;
constexpr int CC   = 3;
constexpr int KK   = 16384;  // power of two -> shift/mask for div/mod
constexpr int HH   = 32;
constexpr int WINR = 2;      // window half-width

// One point per lane; 32 points per wave.
// Layer 1 as H^T = W1^T x P^T via 4x V_WMMA_F32_16X16X4_F32:
//   A (16x4) = W1^T chunk  [hidden x channel]  -- constant per kernel
//   B (4x16) = P^T         [channel x point]   -- built from per-lane pooling
//   D (16x16) = H^T        [hidden x point]
// In D, the hidden index is (j, lane-half) = compile-time constants, and the
// point is the lane's own N column -> layer 2 is lane-local FMAs + ONE
// cross-half shuffle, instead of 64 ds_bpermute butterflies.
__global__ __launch_bounds__(256) void refiner_wmma_kernel(
    const float* __restrict__ x,    // [B,T,C]
    const int*   __restrict__ ids,  // [B,K]
    const float* __restrict__ W1,   // [C,H] row-major: W1[c*H + h]
    const float* __restrict__ b1,   // [H]
    const float* __restrict__ W2,   // [H,1]
    const float* __restrict__ b2,   // [1]
    float*       __restrict__ out)  // [B,K] flat
{
    const int lane    = threadIdx.x & 31;
    const int wave    = threadIdx.x >> 5;
    const int n       = lane & 15;       // N column (point within set)
    const bool hiHalf = lane >= 16;

    // 32 consecutive points per wave; lane l owns point base+l for the gather.
    const long long pointBase =
        ((long long)blockIdx.x * (blockDim.x >> 5) + wave) * 32;
    const long long p = pointBase + lane;

    const int b   = (int)(p >> 14);        // p / KK
    const int k   = (int)(p & (KK - 1));   // p % KK
    const int idx = ids[(size_t)b * KK + k];
    const float* xb = x + (size_t)b * TT * CC;

    // -------- masked-mean pooling over the 5-row window (per-lane gather) ----
    float s0 = 0.f, s1 = 0.f, s2 = 0.f, cnt = 0.f;
#pragma unroll
    for (int d = -WINR; d <= WINR; ++d) {
        int pos  = idx + d;
        float vf = (pos >= 0 && pos < TT) ? 1.f : 0.f;
        int pc   = pos < 0 ? 0 : (pos > TT - 1 ? TT - 1 : pos);
        const float* row = xb + (size_t)pc * CC;   // 12B contiguous
        s0 += row[0] * vf;
        s1 += row[1] * vf;
        s2 += row[2] * vf;
        cnt += vf;
    }
    const float inv = 1.f / cnt;                   // cnt in [3,5], never 0
    const float p0 = s0 * inv, p1 = s1 * inv, p2 = s2 * inv;

    // -------- A operands: W1^T in 16x4 f32 A layout (constants) --------------
    // lanes 0-15: M=h=n,  {K=0: W1[0][h], K=1: W1[1][h]}
    // lanes 16-31: M=h=n, {K=2: W1[2][h], K=3: 0}
    v2f A_lo, A_hi;   // hidden 0..15 and 16..31
    A_lo.x = hiHalf ? W1[2 * HH + n]      : W1[0 * HH + n];
    A_lo.y = hiHalf ? 0.f                 : W1[1 * HH + n];
    A_hi.x = hiHalf ? W1[2 * HH + n + 16] : W1[0 * HH + n + 16];
    A_hi.y = hiHalf ? 0.f                 : W1[1 * HH + n + 16];

    // -------- B operands: P^T in 4x16 f32 B layout ---------------------------
    // Set A = points base..base+15 (owned by lanes 0-15)
    // Set B = points base+16..base+31 (owned by lanes 16-31)
    // lanes 0-15:  {K=0: c0(point n), K=1: c1(point n)}
    // lanes 16-31: {K=2: c2(point n), K=3: 0}
    const float p0_up = __shfl(p0, n + 16, 32);  // set-B c0 for low lanes
    const float p1_up = __shfl(p1, n + 16, 32);  // set-B c1 for low lanes
    const float p2_lo = __shfl(p2, n, 32);       // set-A c2 for high lanes

    v2f B_A, B_B;
    B_A.x = hiHalf ? p2_lo : p0;    B_A.y = hiHalf ? 0.f : p1;
    B_B.x = hiHalf ? p2    : p0_up; B_B.y = hiHalf ? 0.f : p1_up;

    const v8f zero = {};
    // D[m=hidden][n=point]
    v8f D_A_lo = __builtin_amdgcn_wmma_f32_16x16x4_f32(
        false, A_lo, false, B_A, (short)0, zero, false, false);
    v8f D_A_hi = __builtin_amdgcn_wmma_f32_16x16x4_f32(
        false, A_hi, false, B_A, (short)0, zero, false, false);
    v8f D_B_lo = __builtin_amdgcn_wmma_f32_16x16x4_f32(
        false, A_lo, false, B_B, (short)0, zero, false, false);
    v8f D_B_hi = __builtin_amdgcn_wmma_f32_16x16x4_f32(
        false, A_hi, false, B_B, (short)0, zero, false, false);

    // -------- layer 2: bias + ReLU + dot(W2), lane-local ---------------------
    // D VGPR j: lanes 0-15 -> hidden m=j; lanes 16-31 -> hidden m=j+8.
    // Bias/W2 indices are (j, half): scalar loads + one select, no lane gather.
    float partialA = 0.f, partialB = 0.f;
#pragma unroll
    for (int j = 0; j < 8; ++j) {
        const float biaLo = hiHalf ? b1[j + 8]  : b1[j];
        const float w2Lo  = hiHalf ? W2[j + 8]  : W2[j];
        const float biaHi = hiHalf ? b1[j + 24] : b1[j + 16];
        const float w2Hi  = hiHalf ? W2[j + 24] : W2[j + 16];
        partialA += fmaxf(D_A_lo[j] + biaLo, 0.f) * w2Lo
                  + fmaxf(D_A_hi[j] + biaHi, 0.f) * w2Hi;
        partialB += fmaxf(D_B_lo[j] + biaLo, 0.f) * w2Lo
                  + fmaxf(D_B_hi[j] + biaHi, 0.f) * w2Hi;
    }
    // Combine the two hidden halves (lanes n and n+16 hold complementary sums).
    const float rA = partialA + __shfl_xor(partialA, 16, 32);
    const float rB = partialB + __shfl_xor(partialB, 16, 32);
    const float bias2 = b2[0];

    // lanes 0-15 write set-A points, lanes 16-31 write set-B points (coalesced)
    if (!hiHalf) {
        out[pointBase + n] = rA + bias2;
    } else {
        out[pointBase + 16 + n] = rB + bias2;
    }
}

extern "C" void kernel_launch(void* const* d_in, const int* in_sizes, int n_in,
                              void* d_out, int out_size, void* d_ws, size_t ws_size,
                              hipStream_t stream) {
    (void)in_sizes; (void)n_in; (void)out_size; (void)d_ws; (void)ws_size;
    const float* x   = (const float*)d_in[0];
    const int*   ids = (const int*)d_in[1];
    const float* W1  = (const float*)d_in[2];
    const float* b1  = (const float*)d_in[3];
    const float* W2  = (const float*)d_in[4];
    const float* b2  = (const float*)d_in[5];
    float* out = (float*)d_out;

    constexpr int totalPoints   = BB * KK;                    // 1,048,576
    constexpr int threadsPerBlk = 256;                        // 8 waves
    constexpr int pointsPerBlk  = (threadsPerBlk / 32) * 32;  // 256
    constexpr int blocks        = totalPoints / pointsPerBlk; // 4096

    refiner_wmma_kernel<<<blocks, threadsPerBlk, 0, stream>>>(
        x, ids, W1, b1, W2, b2, out);
}